// SRUCell_90632399880534
// MI455X (gfx1250) — compile-verified
//
#include <hip/hip_runtime.h>
#include <math.h>

typedef __attribute__((ext_vector_type(2))) float v2f;
typedef __attribute__((ext_vector_type(4))) float v4f;
typedef __attribute__((ext_vector_type(8))) float v8f;

#define L_SEQ 2048
#define B_DIM 32
#define D_DIM 512
#define N_DIM (3 * D_DIM)        // 1536
#define M_DIM (L_SEQ * B_DIM)    // 65536

#define BM 128
#define BN 128
#define BK 16
#define NIT (D_DIM / BK)         // 32 K-slices
#define LDSS 20                  // padded row stride (floats): mult of 4, spreads banks

// ---------------------------------------------------------------------------
// GEMM: U[M, 1536] = X[M, 512] * W[512, 1536], fp32 via V_WMMA_F32_16X16X4_F32
// LDS double-buffered: one barrier per K-slice; global loads overlap WMMA.
// ---------------------------------------------------------------------------
__global__ __launch_bounds__(256) void sru_gemm_wmma(
    const float* __restrict__ X,
    const float* __restrict__ W,
    float* __restrict__ U)
{
    __shared__ float ldsA[2][BM * LDSS];   // [buf][row m][k]
    __shared__ float ldsB[2][BN * LDSS];   // [buf][col n][k] (transposed)

    const int tid    = threadIdx.x;
    const int wid    = tid >> 5;
    const int lane   = tid & 31;
    const int lane15 = lane & 15;
    const int laneh  = lane >> 4;          // 0/1 -> K-pair select

    const int blockM = blockIdx.y * BM;
    const int blockN = blockIdx.x * BN;

    const int wave_m = wid >> 1;           // 0..3 -> 32 rows each
    const int wave_n = wid & 1;            // 0..1 -> 64 cols each

    v8f acc[2][4];
#pragma unroll
    for (int mi = 0; mi < 2; ++mi)
#pragma unroll
        for (int ni = 0; ni < 4; ++ni)
            acc[mi][ni] = (v8f){0.f,0.f,0.f,0.f,0.f,0.f,0.f,0.f};

    // cooperative-load indices
    const int arow  = tid >> 1;            // 0..127
    const int ahalf = (tid & 1) * 8;       // 0 or 8
    const int brow  = tid & 15;            // k row 0..15
    const int bcol  = (tid >> 4) * 8;      // 0..120

    const float* aptr = X + (size_t)(blockM + arow) * D_DIM + ahalf;
    const float* bptr = W + (size_t)brow * N_DIM + blockN + bcol;

    // ---- preload K-slice 0 into buffer 0 ----
    {
        v4f a0 = *(const v4f*)(aptr);
        v4f a1 = *(const v4f*)(aptr + 4);
        v4f b0 = *(const v4f*)(bptr);
        v4f b1 = *(const v4f*)(bptr + 4);
        *(v4f*)&ldsA[0][arow * LDSS + ahalf]     = a0;
        *(v4f*)&ldsA[0][arow * LDSS + ahalf + 4] = a1;
#pragma unroll
        for (int j = 0; j < 4; ++j) {
            ldsB[0][(bcol + j)     * LDSS + brow] = b0[j];
            ldsB[0][(bcol + 4 + j) * LDSS + brow] = b1[j];
        }
    }
    __syncthreads();

    for (int it = 0; it < NIT; ++it) {
        const int cur = it & 1;
        const int nxt = cur ^ 1;
        const bool has_next = (it + 1 < NIT);

        // issue global loads for next K-slice (in flight during WMMA below)
        v4f a0, a1, b0, b1;
        if (has_next) {
            const int k0n = (it + 1) * BK;
            a0 = *(const v4f*)(aptr + k0n);
            a1 = *(const v4f*)(aptr + k0n + 4);
            b0 = *(const v4f*)(bptr + (size_t)k0n * N_DIM);
            b1 = *(const v4f*)(bptr + (size_t)k0n * N_DIM + 4);
        }

        // compute current K-slice from buffer `cur`
#pragma unroll
        for (int kk = 0; kk < BK; kk += 4) {
            v2f afrag[2], bfrag[4];
#pragma unroll
            for (int mi = 0; mi < 2; ++mi)
                afrag[mi] = *(const v2f*)&ldsA[cur][(wave_m * 32 + mi * 16 + lane15) * LDSS
                                                    + kk + laneh * 2];
#pragma unroll
            for (int ni = 0; ni < 4; ++ni)
                bfrag[ni] = *(const v2f*)&ldsB[cur][(wave_n * 64 + ni * 16 + lane15) * LDSS
                                                    + kk + laneh * 2];
#pragma unroll
            for (int mi = 0; mi < 2; ++mi)
#pragma unroll
                for (int ni = 0; ni < 4; ++ni)
                    acc[mi][ni] = __builtin_amdgcn_wmma_f32_16x16x4_f32(
                        false, afrag[mi], false, bfrag[ni],
                        (short)0, acc[mi][ni], false, false);
        }

        // stage next slice into the other buffer (safe: all waves passed the
        // previous barrier, so nobody is still reading buffer `nxt`)
        if (has_next) {
            *(v4f*)&ldsA[nxt][arow * LDSS + ahalf]     = a0;
            *(v4f*)&ldsA[nxt][arow * LDSS + ahalf + 4] = a1;
#pragma unroll
            for (int j = 0; j < 4; ++j) {
                ldsB[nxt][(bcol + j)     * LDSS + brow] = b0[j];
                ldsB[nxt][(bcol + 4 + j) * LDSS + brow] = b1[j];
            }
        }
        __syncthreads();
    }

    // epilogue: VGPR r holds M=r (lanes 0-15) / M=r+8 (lanes 16-31), N=lane15.
    // U is 2x the L2 and consumed exactly once -> non-temporal stores.
#pragma unroll
    for (int mi = 0; mi < 2; ++mi)
#pragma unroll
        for (int ni = 0; ni < 4; ++ni)
#pragma unroll
            for (int r = 0; r < 8; ++r) {
                int row = blockM + wave_m * 32 + mi * 16 + laneh * 8 + r;
                int col = blockN + wave_n * 64 + ni * 16 + lane15;
                __builtin_nontemporal_store(acc[mi][ni][r], &U[(size_t)row * N_DIM + col]);
            }
}

// ---------------------------------------------------------------------------
// Sequential SRU scan: one thread per (b, d).
// Only 512 waves exist chip-wide (~1/SIMD), so latency must be hidden by ILP:
// chunks of 8 timesteps, double-buffered in registers (32 loads in flight
// while the 8 serial sigmoid steps of the previous chunk execute).
// ---------------------------------------------------------------------------
#define CH 8

__global__ __launch_bounds__(256) void sru_scan(
    const float* __restrict__ X,
    const float* __restrict__ U,
    const float* __restrict__ wc,
    const float* __restrict__ bias,
    const float* __restrict__ c0,
    float* __restrict__ H)
{
    const int idx = blockIdx.x * blockDim.x + threadIdx.x;  // 0..16383
    const int d   = idx & (D_DIM - 1);

    const float vf = wc[d];
    const float vr = wc[D_DIM + d];
    const float bf = bias[d];
    const float br = bias[D_DIM + d];
    const float scale_x = 1.7320508075688772f;  // sqrt(1 + exp(0)*2)

    float c = c0[idx];

    const size_t ustep = (size_t)B_DIM * N_DIM;   // U stride per timestep
    const size_t xstep = (size_t)B_DIM * D_DIM;   // X/H stride per timestep

    const float* up = U + (size_t)(idx >> 9) * N_DIM + (size_t)d * 3;
    const float* xp = X + idx;
    float*       hp = H + idx;

    float u0[CH], u1[CH], u2[CH], xt[CH];
    // load chunk 0 (U single-use -> non-temporal)
#pragma unroll
    for (int j = 0; j < CH; ++j) {
        u0[j] = __builtin_nontemporal_load(up + (size_t)j * ustep);
        u1[j] = __builtin_nontemporal_load(up + (size_t)j * ustep + 1);
        u2[j] = __builtin_nontemporal_load(up + (size_t)j * ustep + 2);
        xt[j] = xp[(size_t)j * xstep];
    }

#pragma unroll 1
    for (int ch = 0; ch < L_SEQ / CH; ++ch) {
        const bool has_next = (ch + 1 < L_SEQ / CH);
        float n0[CH], n1[CH], n2[CH], nx[CH];
        // prefetch next chunk: all 32 loads in flight during the serial chain
        if (has_next) {
            const float* upn = up + (size_t)CH * ustep;
            const float* xpn = xp + (size_t)CH * xstep;
#pragma unroll
            for (int j = 0; j < CH; ++j) {
                n0[j] = __builtin_nontemporal_load(upn + (size_t)j * ustep);
                n1[j] = __builtin_nontemporal_load(upn + (size_t)j * ustep + 1);
                n2[j] = __builtin_nontemporal_load(upn + (size_t)j * ustep + 2);
                nx[j] = xpn[(size_t)j * xstep];
            }
        } else {
#pragma unroll
            for (int j = 0; j < CH; ++j) { n0[j] = n1[j] = n2[j] = nx[j] = 0.f; }
        }

        // 8 serial recurrence steps
#pragma unroll
        for (int j = 0; j < CH; ++j) {
            const float cp = c;
            const float f  = 1.0f / (1.0f + __expf(-(u1[j] + vf * cp + bf)));
            c = u0[j] + (cp - u0[j]) * f;
            const float r  = 1.0f / (1.0f + __expf(-(u2[j] + vr * cp + br)));
            const float ht = r * c + (1.0f - r) * xt[j] * scale_x;
            __builtin_nontemporal_store(ht, hp + (size_t)j * xstep);
        }

#pragma unroll
        for (int j = 0; j < CH; ++j) {
            u0[j] = n0[j]; u1[j] = n1[j]; u2[j] = n2[j]; xt[j] = nx[j];
        }
        up += (size_t)CH * ustep;
        xp += (size_t)CH * xstep;
        hp += (size_t)CH * xstep;
    }
}

extern "C" void kernel_launch(void* const* d_in, const int* in_sizes, int n_in,
                              void* d_out, int out_size, void* d_ws, size_t ws_size,
                              hipStream_t stream) {
    const float* x    = (const float*)d_in[0];   // (L, B, D)
    const float* w    = (const float*)d_in[1];   // (D, 3D)
    const float* wc   = (const float*)d_in[2];   // (2D,)
    const float* bias = (const float*)d_in[3];   // (2D,)
    const float* c0   = (const float*)d_in[4];   // (B, D)
    float* out = (float*)d_out;                  // (L, B, D)
    float* U   = (float*)d_ws;                   // (L*B, 3D) = 384 MB scratch

    dim3 gemm_grid(N_DIM / BN, M_DIM / BM);      // (12, 512)
    sru_gemm_wmma<<<gemm_grid, 256, 0, stream>>>(x, w, U);

    sru_scan<<<(B_DIM * D_DIM) / 256, 256, 0, stream>>>(x, U, wc, bias, c0, out);
}